// MedianFilter1D_9474697855635
// MI455X (gfx1250) — compile-verified
//
#include <hip/hip_runtime.h>
#include <hip/hip_bf16.h>

// MedianFilter1D (k=9, replicate pad) for MI455X / gfx1250.
//
// Roofline: 256 MB total traffic -> ~11 us @ 23.3 TB/s. Naive median9 network
// (38 min/max per output) would be VALU-bound; the 3-operand v_min3/v_med3/
// v_max3 decomposition + intra-thread sorted-triple reuse cuts it to ~11.5
// VALU ops/output, balancing compute with HBM.
//
// Data path: per-block LDS tile staged with GLOBAL_LOAD_ASYNC_TO_LDS_B128
// (ASYNCcnt, VGPR-bypassing) when the builtin is available; otherwise plain
// B128 load + ds_store. Each input byte is read from HBM exactly once.

#define MF_TPB    256                       // threads per block (8 wave32)
#define MF_TILE   1024                      // outputs per block (4 per thread)
#define MF_PAD    4                         // (K-1)/2
#define MF_NCHUNK ((MF_TILE + 2 * MF_PAD) / 4)   // 258 float4 chunks

#if defined(__has_builtin)
#if __has_builtin(__builtin_amdgcn_global_load_async_to_lds_b128)
#define MF_HAS_ASYNC_LDS 1
#endif
#if __has_builtin(__builtin_amdgcn_s_wait_asynccnt)
#define MF_HAS_WAIT_ASYNC 1
#endif
#endif

// 16-byte int vector type matching the async-LDS builtin's pointee type
// (clang diagnostic: param is 'int __vector(4) __device__ *').
typedef int mf_v4i __attribute__((ext_vector_type(4)));
typedef __attribute__((address_space(1))) mf_v4i* mf_gptr;  // global (AS1)
typedef __attribute__((address_space(3))) mf_v4i* mf_lptr;  // LDS (AS3)

static __device__ __forceinline__ void mf_wait_async() {
#if defined(MF_HAS_ASYNC_LDS)
#if defined(MF_HAS_WAIT_ASYNC)
    __builtin_amdgcn_s_wait_asynccnt(0);
#else
    asm volatile("s_wait_asynccnt 0" ::: "memory");
#endif
#endif
}

// Stage one 16-byte chunk c of the (tile + halo) window into LDS.
// Chunk c covers global float indices [tileStart - PAD + 4c, +4), 16B aligned.
static __device__ __forceinline__ void mf_stage_chunk(const float* __restrict__ xrow,
                                                      float* tile, int c,
                                                      int tileStart, int L) {
    const int g = tileStart - MF_PAD + 4 * c;
    float* dst = &tile[4 * c];
    if (g >= 0 && g + 3 < L) {
#if defined(MF_HAS_ASYNC_LDS)
        __builtin_amdgcn_global_load_async_to_lds_b128(
            (mf_gptr)(xrow + g), (mf_lptr)dst, /*offset=*/0, /*cpol=*/0);
#else
        *(float4*)dst = *(const float4*)(xrow + g);
#endif
    } else {
        // Edge chunk: scalar loads with replicate clamping.
#pragma unroll
        for (int j = 0; j < 4; ++j) {
            int idx = g + j;
            idx = idx < 0 ? 0 : (idx > L - 1 ? L - 1 : idx);
            dst[j] = xrow[idx];
        }
    }
}

static __device__ __forceinline__ float mf_min3(float a, float b, float c) {
    return fminf(fminf(a, b), c);          // fuses to v_min3_num_f32
}
static __device__ __forceinline__ float mf_max3(float a, float b, float c) {
    return fmaxf(fmaxf(a, b), c);          // fuses to v_max3_num_f32
}
static __device__ __forceinline__ float mf_med3(float a, float b, float c) {
    return __builtin_amdgcn_fmed3f(a, b, c); // v_med3_f32
}

static __device__ __forceinline__ void mf_sort3(float a, float b, float c,
                                                float& lo, float& mi, float& hi) {
    lo = mf_min3(a, b, c);
    mi = mf_med3(a, b, c);
    hi = mf_max3(a, b, c);
}

__global__ __launch_bounds__(MF_TPB) void MedianFilter1D_9474697855635_kernel(
    const float* __restrict__ x, float* __restrict__ y, int L) {
    __shared__ __align__(16) float tile[MF_TILE + 2 * MF_PAD];  // 1032 floats

    const int t = threadIdx.x;
    const int tileStart = blockIdx.x * MF_TILE;
    const size_t row = (size_t)blockIdx.y * (size_t)L;
    const float* xrow = x + row;
    float* yrow = y + row;

    // ---- Stage tile + halo into LDS (258 chunks over 256 threads) ----
    mf_stage_chunk(xrow, tile, t, tileStart, L);
    if (t < MF_NCHUNK - MF_TPB)  // threads 0,1 pick up chunks 256,257
        mf_stage_chunk(xrow, tile, t + MF_TPB, tileStart, L);

    mf_wait_async();     // drain ASYNCcnt before the barrier makes LDS visible
    __syncthreads();

    // ---- Compute 4 consecutive outputs per thread ----
    const int o = tileStart + 4 * t;  // first output index of this thread
    if (o < L) {
        // Window inputs for outputs o..o+3 live at LDS floats [4t, 4t+12).
        float w[12];
        const float4* tp = (const float4*)(&tile[4 * t]);  // 16B aligned
        float4 v0 = tp[0], v1 = tp[1], v2 = tp[2];         // 3x ds_load_b128
        w[0] = v0.x;  w[1] = v0.y;  w[2]  = v0.z;  w[3]  = v0.w;
        w[4] = v1.x;  w[5] = v1.y;  w[6]  = v1.z;  w[7]  = v1.w;
        w[8] = v2.x;  w[9] = v2.y;  w[10] = v2.z;  w[11] = v2.w;

        // Sorted triples at starts 0..9; outputs j and j+3 reuse triples.
        float lo[10], mi[10], hi[10];
#pragma unroll
        for (int s = 0; s < 10; ++s)
            mf_sort3(w[s], w[s + 1], w[s + 2], lo[s], mi[s], hi[s]);

        // median9 of window j = med3(max3(lows), med3(mids), min3(highs))
        float r[4];
#pragma unroll
        for (int j = 0; j < 4; ++j) {
            const float a = mf_max3(lo[j], lo[j + 3], lo[j + 6]);
            const float b = mf_med3(mi[j], mi[j + 3], mi[j + 6]);
            const float c = mf_min3(hi[j], hi[j + 3], hi[j + 6]);
            r[j] = mf_med3(a, b, c);
        }

        if (o + 3 < L) {
            float4 out;
            out.x = r[0]; out.y = r[1]; out.z = r[2]; out.w = r[3];
            *(float4*)(yrow + o) = out;  // global_store_b128, coalesced
        } else {
#pragma unroll
            for (int j = 0; j < 4; ++j)
                if (o + j < L) yrow[o + j] = r[j];
        }
    }
}

extern "C" void kernel_launch(void* const* d_in, const int* in_sizes, int n_in,
                              void* d_out, int out_size, void* d_ws, size_t ws_size,
                              hipStream_t stream) {
    const float* x = (const float*)d_in[0];
    float* y = (float*)d_out;

    const int B = 32;                 // reference: x is [32, 1000000] f32
    const int L = in_sizes[0] / B;    // 1,000,000

    dim3 grid((L + MF_TILE - 1) / MF_TILE, B);
    dim3 block(MF_TPB);
    MedianFilter1D_9474697855635_kernel<<<grid, block, 0, stream>>>(x, y, L);
}